// DCCA_89635967467549
// MI455X (gfx1250) — compile-verified
//
#include <hip/hip_runtime.h>
#include <stdint.h>

// ---------------------------------------------------------------------------
// Fused triple depthwise 3x3 conv + weighted residual + ReLU for MI455X.
//
// out = relu(x ⊛ (a0*delta + a1*w1 + a2*w2 + a3*w3))  -- single 3x3 dwconv
// after algebraic folding (conv is linear), so 9 FMA/element instead of 31.
// Roofline: 268 MB traffic @ 23.3 TB/s -> ~11.5 us floor; compute需 ~55
// TFLOP/s f32, far below VALU peak => pure bandwidth problem.
// Data path: CDNA5 async global->LDS (ASYNCcnt) + VALU f32 sliding stencil.
// ---------------------------------------------------------------------------

#define C_CH   256
#define H_DIM  128
#define W_DIM  128
#define TILE_H 64
#define LROWS  (TILE_H + 2)   // 66 input rows per tile (top/bottom halo)
#define LPITCH 136            // floats per LDS row: 544 B, 16-B aligned
#define THREADS 256

// ---- CDNA5 async global->LDS helpers (builtin if present, else inline asm) --
#if defined(__has_builtin)
#if __has_builtin(__builtin_amdgcn_global_load_async_to_lds_b128)
#define HAVE_ASYNC_B128 1
#endif
#if __has_builtin(__builtin_amdgcn_s_wait_asynccnt)
#define HAVE_WAIT_ASYNC 1
#endif
#endif
#ifndef HAVE_ASYNC_B128
#define HAVE_ASYNC_B128 0
#endif
#ifndef HAVE_WAIT_ASYNC
#define HAVE_WAIT_ASYNC 0
#endif

typedef __attribute__((ext_vector_type(4))) int v4i;
typedef __attribute__((address_space(1))) v4i* gbl_v4i_ptr;
typedef __attribute__((address_space(3))) v4i* lds_v4i_ptr;

__device__ __forceinline__ void async_copy_b128(const float* g, float* l) {
#if HAVE_ASYNC_B128
  float* gnc = const_cast<float*>(g);
  __builtin_amdgcn_global_load_async_to_lds_b128(
      (gbl_v4i_ptr)gnc, (lds_v4i_ptr)l, 0, 0);
#else
  // VDST = LDS byte offset VGPR, VADDR = 64-bit global address VGPR pair.
  unsigned lds_off = (unsigned)(uintptr_t)l;  // low 32 bits = LDS offset
  asm volatile("global_load_async_to_lds_b128 %0, %1, off"
               :
               : "v"(lds_off), "v"(g)
               : "memory");
#endif
}

__device__ __forceinline__ void wait_async0() {
#if HAVE_WAIT_ASYNC
  __builtin_amdgcn_s_wait_asynccnt(0);
#else
  asm volatile("s_wait_asynccnt 0" ::: "memory");
#endif
}

// ---------------------------------------------------------------------------
// Main kernel (defined first so it leads the disassembly): streaming 3x3
// depthwise conv + ReLU.  One 256-thread block per 64x128 tile of a (b,c)
// plane.  66x128 inputs staged in LDS via async b128 loads (16-B aligned via
// 136-float row pitch); 1-col zero halo each side; per-thread vertical
// register sliding window (3 DS loads + 9 FMA per output); NT b32 stores.
// ---------------------------------------------------------------------------
__global__ __launch_bounds__(THREADS) void dwconv_fused(
    const float* __restrict__ x, const float* __restrict__ weff,
    float* __restrict__ out) {
  __shared__ float lds[LROWS * LPITCH];  // 35,904 B

  const int tid   = threadIdx.x;
  const int plane = blockIdx.x >> 1;     // H/TILE_H = 2 tiles per plane
  const int tile  = blockIdx.x & 1;
  const int c     = plane & (C_CH - 1);

  const float* __restrict__ src = x + (size_t)plane * (H_DIM * W_DIM);
  float* __restrict__ dst       = out + (size_t)plane * (H_DIM * W_DIM);
  const int row0 = tile * TILE_H - 1;    // global row held in LDS row 0

  // Kick off async tile load (rows row0..row0+65) as 16-B chunks; zero OOB
  // rows with DS stores on the fly.
  for (int i = tid; i < LROWS * (W_DIM / 4); i += THREADS) {
    int r  = i >> 5;        // LDS row 0..65
    int ch = i & 31;        // 16-B chunk within row
    float* l = &lds[r * LPITCH + 4 + ch * 4];   // 16-B aligned
    int gr = row0 + r;
    if ((unsigned)gr < (unsigned)H_DIM) {
      async_copy_b128(src + (size_t)gr * W_DIM + ch * 4, l);
    } else {
      l[0] = 0.0f; l[1] = 0.0f; l[2] = 0.0f; l[3] = 0.0f;
    }
  }

  // Zero left/right halo columns (data at cols 4..131; halos at 3 and 132)
  // while the async transfers are in flight.
  for (int i = tid; i < LROWS * 2; i += THREADS) {
    int r = i >> 1;
    lds[r * LPITCH + 3 + (i & 1) * 129] = 0.0f;
  }

  wait_async0();
  __syncthreads();

  // Per-channel effective 3x3 kernel: block-uniform -> scalar (SGPR) loads.
  const float* __restrict__ wp = weff + c * 9;
  const float k00 = wp[0], k01 = wp[1], k02 = wp[2];
  const float k10 = wp[3], k11 = wp[4], k12 = wp[5];
  const float k20 = wp[6], k21 = wp[7], k22 = wp[8];

  // Thread -> (column w, row half).  Each thread: 32 output rows, sliding
  // 3x3 register window.  Output row g = tile*64 + half*32 + r lives at LDS
  // row (half*32 + r + 1); taps w-1..w+1 are LDS cols 3+w .. 5+w.
  const int w    = tid & (W_DIM - 1);
  const int half = tid >> 7;

  const float* lp = &lds[(half * 32) * LPITCH + 3 + w];
  float t0 = lp[0], t1 = lp[1], t2 = lp[2];
  lp += LPITCH;
  float m0 = lp[0], m1 = lp[1], m2 = lp[2];

  float* drow = dst + (size_t)(tile * TILE_H + half * 32) * W_DIM + w;
#pragma unroll
  for (int r = 0; r < 32; ++r) {
    lp += LPITCH;
    float b0 = lp[0], b1 = lp[1], b2 = lp[2];
    float acc = t0 * k00;
    acc = fmaf(t1, k01, acc);
    acc = fmaf(t2, k02, acc);
    acc = fmaf(m0, k10, acc);
    acc = fmaf(m1, k11, acc);
    acc = fmaf(m2, k12, acc);
    acc = fmaf(b0, k20, acc);
    acc = fmaf(b1, k21, acc);
    acc = fmaf(b2, k22, acc);
    __builtin_nontemporal_store(fmaxf(acc, 0.0f), drow);
    drow += W_DIM;
    t0 = m0; t1 = m1; t2 = m2;
    m0 = b0; m1 = b1; m2 = b2;
  }
}

// ---------------------------------------------------------------------------
// Prep kernel: fold the three 3x3 kernels + residual scale into one
// effective per-channel 3x3 kernel.
//   weff[c*9+k] = a1*w1 + a2*w2 + a3*w3   (+a0 at center tap k==4)
// ---------------------------------------------------------------------------
__global__ void weff_prep(const float* __restrict__ a,
                          const float* __restrict__ w1,
                          const float* __restrict__ w2,
                          const float* __restrict__ w3,
                          float* __restrict__ weff) {
  int i = blockIdx.x * blockDim.x + threadIdx.x;
  if (i < C_CH * 9) {
    float v = a[1] * w1[i];
    v = fmaf(a[2], w2[i], v);
    v = fmaf(a[3], w3[i], v);
    if ((i % 9) == 4) v += a[0];   // identity (residual) folded into center tap
    weff[i] = v;
  }
}

// ---------------------------------------------------------------------------
extern "C" void kernel_launch(void* const* d_in, const int* in_sizes, int n_in,
                              void* d_out, int out_size, void* d_ws,
                              size_t ws_size, hipStream_t stream) {
  const float* x  = (const float*)d_in[0];
  const float* a  = (const float*)d_in[1];
  const float* w1 = (const float*)d_in[2];
  const float* w2 = (const float*)d_in[3];
  const float* w3 = (const float*)d_in[4];
  float* out  = (float*)d_out;
  float* weff = (float*)d_ws;   // 256*9 floats = 9216 B of scratch

  weff_prep<<<(C_CH * 9 + 255) / 256, 256, 0, stream>>>(a, w1, w2, w3, weff);

  const int planes  = out_size / (H_DIM * W_DIM);        // B*C = 2048
  const int nblocks = planes * (H_DIM / TILE_H);         // 4096
  dwconv_fused<<<nblocks, THREADS, 0, stream>>>(x, weff, out);
}